// ChemGNNModel_28303834481484
// MI455X (gfx1250) — compile-verified
//
#include <hip/hip_runtime.h>
#include <hip/hip_bf16.h>
#include <math.h>

typedef float v2f __attribute__((ext_vector_type(2)));
typedef float v8f __attribute__((ext_vector_type(8)));

#define NNODES 10000
#define NEDGES 160000
#define NGRAPH 128
#define DH     100     // hidden width
#define FP1    112     // padded hidden (7 x 16)
#define FP0    16      // padded layer-0 fan-in (11 -> 16)

// ---------- tiny prep kernels ----------------------------------------------
__global__ void k_softmax5(const float* __restrict__ w, float* __restrict__ out) {
  if (threadIdx.x == 0) {
    float m = w[0];
    for (int i = 1; i < 5; ++i) m = fmaxf(m, w[i]);
    float e[5], s = 0.f;
    for (int i = 0; i < 5; ++i) { e[i] = __expf(w[i] - m); s += e[i]; }
    for (int i = 0; i < 5; ++i) out[i] = e[i] / s;
  }
}

__global__ void k_pad_x0(const float* __restrict__ x, float* __restrict__ xp, int tot) {
  int i = blockIdx.x * blockDim.x + threadIdx.x;
  if (i >= tot) return;
  int r = i >> 4, c = i & 15;
  xp[i] = (c < 11) ? x[r * 11 + c] : 0.f;
}

// pre_W [3*fin, fin] row-major -> padded Wdst/Wsrc [Fp, Fp]
__global__ void k_pack_pre(const float* __restrict__ preW, float* __restrict__ Wd,
                           float* __restrict__ Ws, int fin, int Fp) {
  int i = blockIdx.x * blockDim.x + threadIdx.x;
  if (i >= Fp * Fp) return;
  int k = i / Fp, j = i - k * Fp;
  float vd = 0.f, vs = 0.f;
  if (k < fin && j < fin) {
    vd = preW[k * fin + j];
    vs = preW[(fin + k) * fin + j];
  }
  Wd[i] = vd; Ws[i] = vs;
}

// T[a][j] = (edge_emb[a]·enc_W + enc_b)·W_e[:,j] + pre_b[j]   (20 x Fp table)
__global__ void k_build_T(const float* __restrict__ eemb, const float* __restrict__ encW,
                          const float* __restrict__ encB, const float* __restrict__ preW,
                          const float* __restrict__ preB, float* __restrict__ T,
                          int fin, int Fp) {
  int i = blockIdx.x * blockDim.x + threadIdx.x;
  if (i >= 20 * Fp) return;
  int a = i / Fp, j = i - a * Fp;
  float acc = 0.f;
  if (j < fin) {
    for (int k = 0; k < fin; ++k) {
      float ek = encB[k];
      for (int q = 0; q < 10; ++q) ek += eemb[a * 10 + q] * encW[q * fin + k];
      acc += ek * preW[(2 * fin + k) * fin + j];
    }
    acc += preB[j];
  }
  T[a * Fp + j] = acc;
}

// Wc[6*Fp, FP1] = pad( post_W[6*fin,100] @ lin_W[100,100] )
__global__ void k_fold_post(const float* __restrict__ postW, const float* __restrict__ linW,
                            float* __restrict__ Wc, int fin, int Fp) {
  int i = blockIdx.x * blockDim.x + threadIdx.x;
  if (i >= 6 * Fp * FP1) return;
  int kk = i / FP1, j = i - kk * FP1;
  int blk = kk / Fp, off = kk - blk * Fp;
  float acc = 0.f;
  if (off < fin && j < DH) {
    int srow = blk * fin + off;
    for (int t = 0; t < DH; ++t) acc += postW[srow * DH + t] * linW[t * DH + j];
  }
  Wc[i] = acc;
}

__global__ void k_fold_bias(const float* __restrict__ postB, const float* __restrict__ linW,
                            const float* __restrict__ linB, float* __restrict__ bc) {
  int j = blockIdx.x * blockDim.x + threadIdx.x;
  if (j >= FP1) return;
  float acc = 0.f;
  if (j < DH) {
    for (int t = 0; t < DH; ++t) acc += postB[t] * linW[t * DH + j];
    acc += linB[j];
  }
  bc[j] = acc;
}

// ---------- f32 WMMA GEMM: C[M,Nc] = A[M,K] @ W[K,Nc] (+bias) ---------------
// grid = (tiles_M / wavesPerBlock, tiles_N); one wave per 16x16 tile.
// All waves of a block share blockIdx.y (same 16-col W strip) -> B loads hit L0.
// V_WMMA_F32_16X16X4_F32 fragment layouts:
//   A (16x4): lanes 0-15 -> K=k,k+1 ; lanes 16-31 -> K=k+2,k+3 (row = lane&15)
//   B (4x16): same K split, column = lane&15
//   C/D     : VGPR r -> row r + 8*(lane>>4), col = lane&15
__global__ void k_wmma_gemm(const float* __restrict__ A, const float* __restrict__ W,
                            const float* __restrict__ bias, float* __restrict__ C,
                            int M, int K, int Nc) {
  int tn = blockIdx.y;
  int tm = blockIdx.x * (blockDim.x >> 5) + (threadIdx.x >> 5);
  if (tm >= (M >> 4)) return;
  int lane = threadIdx.x & 31;
  int half = lane >> 4, l16 = lane & 15;

  const float* arow = A + (size_t)(tm * 16 + l16) * K;
  const float* wcol = W + tn * 16 + l16;

  v8f acc = {0.f, 0.f, 0.f, 0.f, 0.f, 0.f, 0.f, 0.f};
  for (int k = 0; k < K; k += 4) {
    if ((k & 63) == 0) __builtin_prefetch(arow + k + 64, 0, 1);  // global_prefetch_b8
    int ka = k + 2 * half;
    v2f a, b;
    a.x = arow[ka];
    a.y = arow[ka + 1];
    b.x = wcol[(size_t)ka * Nc];
    b.y = wcol[(size_t)(ka + 1) * Nc];
    acc = __builtin_amdgcn_wmma_f32_16x16x4_f32(false, a, false, b,
                                                (short)0, acc, false, false);
  }
  float bv = bias ? bias[tn * 16 + l16] : 0.f;
  float* cbase = C + (size_t)(tm * 16 + 8 * half) * Nc + tn * 16 + l16;
#pragma unroll
  for (int r = 0; r < 8; ++r) cbase[(size_t)r * Nc] = acc[r] + bv;
}

// ---------- CSR build (once per launch; edge list is layer-invariant) -------
__global__ void k_zero_i(int* __restrict__ p, int n) {
  int i = blockIdx.x * blockDim.x + threadIdx.x;
  if (i < n) p[i] = 0;
}

__global__ void k_count(const int* __restrict__ eidx, int* __restrict__ cnt, int E) {
  int e = blockIdx.x * blockDim.x + threadIdx.x;
  if (e >= E) return;
  atomicAdd(&cnt[eidx[E + e]], 1);
}

// single-block exclusive scan: rowptr[0..n] from cnt[0..n-1]
__global__ void k_scan(const int* __restrict__ cnt, int* __restrict__ rowptr, int n) {
  __shared__ int sh[256];
  __shared__ int carry;
  if (threadIdx.x == 0) carry = 0;
  __syncthreads();
  for (int base = 0; base < n; base += 256) {
    int i = base + (int)threadIdx.x;
    int v = (i < n) ? cnt[i] : 0;
    sh[threadIdx.x] = v;
    __syncthreads();
    for (int off = 1; off < 256; off <<= 1) {
      int t = ((int)threadIdx.x >= off) ? sh[threadIdx.x - off] : 0;
      __syncthreads();
      sh[threadIdx.x] += t;
      __syncthreads();
    }
    int incl = sh[threadIdx.x];
    if (i < n) rowptr[i] = carry + incl - v;
    __syncthreads();
    if (threadIdx.x == 255) carry += incl;
    __syncthreads();
  }
  if (threadIdx.x == 0) rowptr[n] = carry;
}

__global__ void k_fill_csr(const int* __restrict__ eidx, const int* __restrict__ eattr,
                           const int* __restrict__ rowptr, int* __restrict__ cursor,
                           int* __restrict__ s_src, int* __restrict__ s_attr, int E) {
  int e = blockIdx.x * blockDim.x + threadIdx.x;
  if (e >= E) return;
  int d = eidx[E + e];
  int slot = rowptr[d] + atomicAdd(&cursor[d], 1);
  s_src[slot]  = eidx[e];
  s_attr[slot] = eattr[e];
}

// ---------- fused node-major aggregation + Y build (no atomics) -------------
// Y[node] = [ x | w0*sum | w1*mean | w2*min | w3*max | w4*std ]
// NPB nodes per block; FP threads per node (thread = one column).
template <int FP, int NPB>
__global__ void k_aggregate_Y(const int* __restrict__ rowptr, const int* __restrict__ s_src,
                              const int* __restrict__ s_attr, const float* __restrict__ Ad,
                              const float* __restrict__ Bs, const float* __restrict__ T,
                              const float* __restrict__ X, const float* __restrict__ w,
                              float* __restrict__ Y) {
  int node = blockIdx.x * NPB + (int)threadIdx.x / FP;
  int c = (int)threadIdx.x % FP;
  if (node >= NNODES) return;
  int beg = rowptr[node], end = rowptr[node + 1];
  float a = Ad[(size_t)node * FP + c];
  float s = 0.f, s2 = 0.f;
  float mn = INFINITY, mx = -INFINITY;
  for (int j = beg; j < end; ++j) {
    int sr = s_src[j];             // broadcast load (uniform across node's threads)
    int at = s_attr[j];
    float m = a + Bs[(size_t)sr * FP + c] + T[at * FP + c];
    s += m;
    s2 += m * m;
    mn = fminf(mn, m);
    mx = fmaxf(mx, m);
  }
  float deg = (float)(end - beg);
  float cc = fmaxf(deg, 1.f);
  float mean = s / cc;
  float var = s2 / cc - mean * mean;
  float sd = sqrtf(fmaxf(var, 0.f) + 1e-5f);
  if (deg <= 0.f) { mn = 0.f; mx = 0.f; }
  float* y = Y + (size_t)node * (6 * FP) + c;
  y[0]      = X[(size_t)node * FP + c];
  y[FP]     = w[0] * s;
  y[2 * FP] = w[1] * mean;
  y[3 * FP] = w[2] * mn;
  y[4 * FP] = w[3] * mx;
  y[5 * FP] = w[4] * sd;
}

// ---------- batchnorm -------------------------------------------------------
__global__ void k_bn_stats(const float* __restrict__ H, float* __restrict__ mu,
                           float* __restrict__ var, int n) {
  int c = blockIdx.x;  // one block per column, FP1 blocks
  float s = 0.f, s2 = 0.f;
  for (int r = threadIdx.x; r < n; r += blockDim.x) {
    float v = H[(size_t)r * FP1 + c];
    s += v; s2 += v * v;
  }
  __shared__ float sh[256], sh2[256];
  sh[threadIdx.x] = s; sh2[threadIdx.x] = s2;
  __syncthreads();
  for (int off = blockDim.x >> 1; off > 0; off >>= 1) {
    if ((int)threadIdx.x < off) {
      sh[threadIdx.x]  += sh[threadIdx.x + off];
      sh2[threadIdx.x] += sh2[threadIdx.x + off];
    }
    __syncthreads();
  }
  if (threadIdx.x == 0) {
    float m = sh[0] / n;
    mu[c] = m;
    var[c] = sh2[0] / n - m * m;
  }
}

__global__ void k_bn_relu(const float* __restrict__ H, const float* __restrict__ mu,
                          const float* __restrict__ var, const float* __restrict__ g,
                          const float* __restrict__ b, float* __restrict__ Xo, int n) {
  int i = blockIdx.x * blockDim.x + threadIdx.x;
  if (i >= n * FP1) return;
  int c = i % FP1;
  float o = 0.f;
  if (c < DH) {
    float inv = rsqrtf(var[c] + 1e-5f);
    o = fmaxf((H[i] - mu[c]) * inv * g[c] + b[c], 0.f);
  }
  Xo[i] = o;
}

// ---------- final MLP + graph pooling ---------------------------------------
__global__ void k_mlp(const float* __restrict__ X, const float* __restrict__ W,
                      const float* __restrict__ bias, float* __restrict__ O,
                      int n, int Kin, int Kstride, int Nout) {
  int i = blockIdx.x * blockDim.x + threadIdx.x;
  if (i >= n * Nout) return;
  int r = i / Nout, j = i - r * Nout;
  float acc = bias[j];
  for (int k = 0; k < Kin; ++k) acc += X[(size_t)r * Kstride + k] * W[k * Nout + j];
  O[i] = fmaxf(acc, 0.f);
}

__global__ void k_zero_f(float* __restrict__ p, int n) {
  int i = blockIdx.x * blockDim.x + threadIdx.x;
  if (i < n) p[i] = 0.f;
}

__global__ void k_mlp3_scatter(const float* __restrict__ H2, const float* __restrict__ W3,
                               const float* __restrict__ b3, const int* __restrict__ batch,
                               float* __restrict__ out, int n) {
  int r = blockIdx.x * blockDim.x + threadIdx.x;
  if (r >= n) return;
  float acc = b3[0];
  for (int k = 0; k < 20; ++k) acc += H2[r * 20 + k] * W3[k];
  atomicAdd(&out[batch[r]], acc);
}

// ---------- host orchestration ----------------------------------------------
static inline int cdiv(long long a, int b) { return (int)((a + b - 1) / b); }

extern "C" void kernel_launch(void* const* d_in, const int* in_sizes, int n_in,
                              void* d_out, int out_size, void* d_ws, size_t ws_size,
                              hipStream_t stream) {
  (void)in_sizes; (void)n_in; (void)out_size; (void)ws_size;

  // pytree-leaf order: x, edge_index, edge_attr, batch, then params with dict
  // keys sorted: agg_w, edge_emb, layers[0..9]{bn_b,bn_g,enc_W,enc_b,lin_W,
  // lin_b,post_W,post_b,pre_W,pre_b}, mlp{W1,W2,W3,b1,b2,b3}
  const float* x_in  = (const float*)d_in[0];
  const int*   eidx  = (const int*)d_in[1];
  const int*   eattr = (const int*)d_in[2];
  const int*   batch = (const int*)d_in[3];
  const float* agg_w = (const float*)d_in[4];
  const float* eemb  = (const float*)d_in[5];
  const int LBASE = 6;
  const float* W1 = (const float*)d_in[LBASE + 100 + 0];
  const float* W2 = (const float*)d_in[LBASE + 100 + 1];
  const float* W3 = (const float*)d_in[LBASE + 100 + 2];
  const float* b1 = (const float*)d_in[LBASE + 100 + 3];
  const float* b2 = (const float*)d_in[LBASE + 100 + 4];
  const float* b3 = (const float*)d_in[LBASE + 100 + 5];

  // workspace bump allocator (256B aligned)
  char* wp = (char*)d_ws;
  auto alloc = [&](size_t bytes) -> void* {
    char* r = wp;
    wp += (bytes + 255) & ~(size_t)255;
    return (void*)r;
  };
  float* X0     = (float*)alloc((size_t)NNODES * FP0 * 4);
  float* Xa     = (float*)alloc((size_t)NNODES * FP1 * 4);
  float* Xb     = (float*)alloc((size_t)NNODES * FP1 * 4);
  float* Abuf   = (float*)alloc((size_t)NNODES * FP1 * 4);
  float* Bbuf   = (float*)alloc((size_t)NNODES * FP1 * 4);
  float* Y      = (float*)alloc((size_t)NNODES * 6 * FP1 * 4);
  float* H      = (float*)alloc((size_t)NNODES * FP1 * 4);
  float* T      = (float*)alloc((size_t)20 * FP1 * 4);
  float* Wdst   = (float*)alloc((size_t)FP1 * FP1 * 4);
  float* Wsrc   = (float*)alloc((size_t)FP1 * FP1 * 4);
  float* Wc     = (float*)alloc((size_t)6 * FP1 * FP1 * 4);
  float* bc     = (float*)alloc((size_t)FP1 * 4);
  float* w5     = (float*)alloc(5 * 4);
  float* mu     = (float*)alloc((size_t)FP1 * 4);
  float* var    = (float*)alloc((size_t)FP1 * 4);
  float* h1     = (float*)alloc((size_t)NNODES * 50 * 4);
  float* h2     = (float*)alloc((size_t)NNODES * 20 * 4);
  int*   cntI   = (int*)alloc((size_t)NNODES * 4);
  int*   rowptr = (int*)alloc((size_t)(NNODES + 1) * 4);
  int*   cursor = (int*)alloc((size_t)NNODES * 4);
  int*   s_src  = (int*)alloc((size_t)NEDGES * 4);
  int*   s_attr = (int*)alloc((size_t)NEDGES * 4);

  // ---- one-time prep: softmax weights, padded x0, dst-CSR --------------------
  k_softmax5<<<1, 32, 0, stream>>>(agg_w, w5);
  k_pad_x0<<<cdiv((long long)NNODES * FP0, 256), 256, 0, stream>>>(x_in, X0, NNODES * FP0);
  k_zero_i<<<cdiv(NNODES, 256), 256, 0, stream>>>(cntI, NNODES);
  k_zero_i<<<cdiv(NNODES, 256), 256, 0, stream>>>(cursor, NNODES);
  k_count<<<cdiv(NEDGES, 256), 256, 0, stream>>>(eidx, cntI, NEDGES);
  k_scan<<<1, 256, 0, stream>>>(cntI, rowptr, NNODES);
  k_fill_csr<<<cdiv(NEDGES, 256), 256, 0, stream>>>(eidx, eattr, rowptr, cursor,
                                                    s_src, s_attr, NEDGES);

  const float* Xcur = X0;
  float* Xnext = Xa;

  for (int l = 0; l < 10; ++l) {
    const int fin = (l == 0) ? 11 : DH;
    const int Fp  = (l == 0) ? FP0 : FP1;
    const float* bn_b  = (const float*)d_in[LBASE + 10 * l + 0];
    const float* bn_g  = (const float*)d_in[LBASE + 10 * l + 1];
    const float* encW  = (const float*)d_in[LBASE + 10 * l + 2];
    const float* encB  = (const float*)d_in[LBASE + 10 * l + 3];
    const float* linW  = (const float*)d_in[LBASE + 10 * l + 4];
    const float* linB  = (const float*)d_in[LBASE + 10 * l + 5];
    const float* postW = (const float*)d_in[LBASE + 10 * l + 6];
    const float* postB = (const float*)d_in[LBASE + 10 * l + 7];
    const float* preW  = (const float*)d_in[LBASE + 10 * l + 8];
    const float* preB  = (const float*)d_in[LBASE + 10 * l + 9];

    // weight prep (tiny)
    k_pack_pre<<<cdiv(Fp * Fp, 256), 256, 0, stream>>>(preW, Wdst, Wsrc, fin, Fp);
    k_build_T<<<cdiv(20 * Fp, 64), 64, 0, stream>>>(eemb, encW, encB, preW, preB, T, fin, Fp);
    k_fold_post<<<cdiv(6 * Fp * FP1, 256), 256, 0, stream>>>(postW, linW, Wc, fin, Fp);
    k_fold_bias<<<1, FP1, 0, stream>>>(postB, linW, linB, bc);

    // A = X @ Wdst, B = X @ Wsrc (WMMA f32)
    {
      dim3 grid(cdiv(NNODES / 16, 8), Fp / 16);
      k_wmma_gemm<<<grid, 256, 0, stream>>>(Xcur, Wdst, nullptr, Abuf, NNODES, Fp, Fp);
      k_wmma_gemm<<<grid, 256, 0, stream>>>(Xcur, Wsrc, nullptr, Bbuf, NNODES, Fp, Fp);
    }

    // fused CSR aggregation + Y build (atomic-free)
    if (l == 0)
      k_aggregate_Y<FP0, 8><<<cdiv(NNODES, 8), 128, 0, stream>>>(
          rowptr, s_src, s_attr, Abuf, Bbuf, T, Xcur, w5, Y);
    else
      k_aggregate_Y<FP1, 1><<<NNODES, FP1, 0, stream>>>(
          rowptr, s_src, s_attr, Abuf, Bbuf, T, Xcur, w5, Y);

    // H = Y @ Wc + bc (WMMA f32)
    {
      dim3 grid(cdiv(NNODES / 16, 8), FP1 / 16);
      k_wmma_gemm<<<grid, 256, 0, stream>>>(Y, Wc, bc, H, NNODES, 6 * Fp, FP1);
    }

    // batchnorm + relu -> next X
    k_bn_stats<<<FP1, 256, 0, stream>>>(H, mu, var, NNODES);
    k_bn_relu<<<cdiv((long long)NNODES * FP1, 256), 256, 0, stream>>>(
        H, mu, var, bn_g, bn_b, Xnext, NNODES);

    Xcur = Xnext;
    Xnext = (Xnext == Xa) ? Xb : Xa;
  }

  // final MLP 100 -> 50 -> 20 -> 1, then segment-sum over batch
  k_mlp<<<cdiv((long long)NNODES * 50, 256), 256, 0, stream>>>(Xcur, W1, b1, h1,
                                                               NNODES, DH, FP1, 50);
  k_mlp<<<cdiv((long long)NNODES * 20, 256), 256, 0, stream>>>(h1, W2, b2, h2,
                                                               NNODES, 50, 50, 20);
  k_zero_f<<<1, 128, 0, stream>>>((float*)d_out, NGRAPH);
  k_mlp3_scatter<<<cdiv(NNODES, 256), 256, 0, stream>>>(h2, W3, b3, batch,
                                                        (float*)d_out, NNODES);
}